// GIN_28226525069445
// MI455X (gfx1250) — compile-verified
//
#include <hip/hip_runtime.h>
#include <hip/hip_bf16.h>

typedef __attribute__((ext_vector_type(2))) float v2f;
typedef __attribute__((ext_vector_type(8))) float v8f;

#define NFEAT 128
#define BN_EPS 1e-5f

// ---------------------------------------------------------------------------
// Kernel 1: h0 = x  (scatter target starts at x, since h = (1+0)*x + agg),
//           and zero the BN stat accumulators (sum[128], sumsq[128]).
// ---------------------------------------------------------------------------
__global__ __launch_bounds__(256) void k_init(const float4* __restrict__ x4,
                                              float4* __restrict__ h04,
                                              float* __restrict__ stats,
                                              int n4) {
  int i = blockIdx.x * 256 + threadIdx.x;
  if (i < n4) h04[i] = x4[i];
  if (blockIdx.x == 0 && threadIdx.x < 256) stats[threadIdx.x] = 0.0f;
}

// ---------------------------------------------------------------------------
// Kernel 2: edge scatter-add. One wave (32 lanes) per edge; each lane moves
// 4 consecutive floats (float4 gather + 4 global f32 atomics).
// ---------------------------------------------------------------------------
__global__ __launch_bounds__(256) void k_scatter(const int* __restrict__ src,
                                                 const int* __restrict__ dst,
                                                 const float* __restrict__ x,
                                                 float* __restrict__ h0,
                                                 int n_edges) {
  int e = blockIdx.x * 8 + (threadIdx.x >> 5);
  if (e >= n_edges) return;
  int lane = threadIdx.x & 31;
  long long s = (long long)src[e];
  long long d = (long long)dst[e];
  float4 v = ((const float4*)(x + s * NFEAT))[lane];
  float* o = h0 + d * NFEAT + lane * 4;
  atomicAdd(o + 0, v.x);
  atomicAdd(o + 1, v.y);
  atomicAdd(o + 2, v.z);
  atomicAdd(o + 3, v.w);
}

// ---------------------------------------------------------------------------
// WMMA GEMM: C[M x 128] = A[M x 128] @ W[128 x 128] + bias, optional ReLU.
// MODE 0: GEMM1 (plain A, bias + ReLU epilogue)
// MODE 1: GEMM2 (A affine-transformed per K-column: a*scale[k]+shift[k],
//                bias epilogue, no ReLU)
// Block = 256 threads (8 waves). Block b owns rows [16b, 16b+16); wave w owns
// columns [16w, 16w+16). Inner loop: 32 x v_wmma_f32_16x16x4_f32.
// LDS padded to stride 132 dwords -> conflict-free fragment reads.
// ---------------------------------------------------------------------------
template <int MODE>
__global__ __launch_bounds__(256) void k_gemm(const float* __restrict__ A,
                                              const float* __restrict__ W,
                                              const float* __restrict__ bias,
                                              const float* __restrict__ scale,
                                              const float* __restrict__ shift,
                                              float* __restrict__ out,
                                              int M) {
  __shared__ float As[16][132];
  __shared__ float Bs[128][132];
  __shared__ float sBias[128];
  __shared__ float sScale[128];
  __shared__ float sShift[128];

  const int tid = threadIdx.x;
  const int row0 = blockIdx.x * 16;
  if (row0 >= M) return;

  // Stage W (128x128) into LDS, coalesced float4.
  {
    const float4* W4 = (const float4*)W;
    for (int idx = tid; idx < 128 * 32; idx += 256) {
      int r = idx >> 5;         // row of W (k)
      int c4 = idx & 31;        // float4 column
      float4 v = W4[r * 32 + c4];
      Bs[r][c4 * 4 + 0] = v.x;
      Bs[r][c4 * 4 + 1] = v.y;
      Bs[r][c4 * 4 + 2] = v.z;
      Bs[r][c4 * 4 + 3] = v.w;
    }
  }
  if (tid < 128) {
    sBias[tid] = bias[tid];
    if (MODE == 1) {
      sScale[tid] = scale[tid];
      sShift[tid] = shift[tid];
    }
  }
  __syncthreads();

  // Stage A tile (16 x 128), applying BN affine in MODE 1.
  for (int idx = tid; idx < 16 * 128; idx += 256) {
    int r = idx >> 7;
    int c = idx & 127;
    float v = A[(long long)(row0 + r) * NFEAT + c];
    if (MODE == 1) v = v * sScale[c] + sShift[c];
    As[r][c] = v;
  }
  __syncthreads();

  const int wave = tid >> 5;        // 0..7  -> N tile
  const int lane = tid & 31;
  const int l16 = lane & 15;
  const int hi = lane >> 4;         // 0 or 1
  const int col0 = wave * 16;

  v8f c = {};
  #pragma unroll
  for (int k0 = 0; k0 < 128; k0 += 4) {
    const int ka = k0 + hi * 2;
    v2f a;
    a.x = As[l16][ka];
    a.y = As[l16][ka + 1];
    v2f b;
    b.x = Bs[ka][col0 + l16];
    b.y = Bs[ka + 1][col0 + l16];
    c = __builtin_amdgcn_wmma_f32_16x16x4_f32(false, a, false, b,
                                              (short)0, c, false, false);
  }

  // Epilogue: D[m][n] layout -> lane<16: (M=v, N=l16); lane>=16: (M=v+8, N=l16)
  const int colg = col0 + l16;
  const float bv = sBias[colg];
  #pragma unroll
  for (int v = 0; v < 8; ++v) {
    int m = v + hi * 8;
    float r = c[v] + bv;
    if (MODE == 0) r = fmaxf(r, 0.0f);
    out[(long long)(row0 + m) * NFEAT + colg] = r;
  }
}

// ---------------------------------------------------------------------------
// Kernel 4: per-feature sum / sumsq of h1. Each thread's feature is fixed
// (tid & 127) because stride is a multiple of 128. LDS reduce -> 1 atomic
// per feature per block.
// ---------------------------------------------------------------------------
__global__ __launch_bounds__(256) void k_stats(const float* __restrict__ h1,
                                               float* __restrict__ sum,
                                               float* __restrict__ sumsq,
                                               int total) {
  __shared__ float ls[256];
  __shared__ float ls2[256];
  float s = 0.0f, s2 = 0.0f;
  for (int i = blockIdx.x * 256 + threadIdx.x; i < total;
       i += gridDim.x * 256) {
    float v = h1[i];
    s += v;
    s2 += v * v;
  }
  ls[threadIdx.x] = s;
  ls2[threadIdx.x] = s2;
  __syncthreads();
  if (threadIdx.x < 128) {
    float a = ls[threadIdx.x] + ls[threadIdx.x + 128];
    float b = ls2[threadIdx.x] + ls2[threadIdx.x + 128];
    atomicAdd(&sum[threadIdx.x], a);
    atomicAdd(&sumsq[threadIdx.x], b);
  }
}

// ---------------------------------------------------------------------------
// Kernel 5: fold BN stats + gamma/beta into per-feature scale/shift.
// ---------------------------------------------------------------------------
__global__ void k_bnfinal(const float* __restrict__ sum,
                          const float* __restrict__ sumsq,
                          const float* __restrict__ gamma,
                          const float* __restrict__ beta,
                          float* __restrict__ scale,
                          float* __restrict__ shift, float invN) {
  int f = threadIdx.x;
  if (f >= 128) return;
  float mean = sum[f] * invN;
  float var = sumsq[f] * invN - mean * mean;
  float rstd = rsqrtf(var + BN_EPS);
  float sc = rstd * gamma[f];
  scale[f] = sc;
  shift[f] = beta[f] - mean * sc;
}

// ---------------------------------------------------------------------------
// Launch
// ---------------------------------------------------------------------------
extern "C" void kernel_launch(void* const* d_in, const int* in_sizes, int n_in,
                              void* d_out, int out_size, void* d_ws,
                              size_t ws_size, hipStream_t stream) {
  const float* x = (const float*)d_in[0];
  const int* edge = (const int*)d_in[1];
  const float* W1 = (const float*)d_in[2];
  const float* b1 = (const float*)d_in[3];
  const float* gamma = (const float*)d_in[4];
  const float* beta = (const float*)d_in[5];
  const float* W2 = (const float*)d_in[6];
  const float* b2 = (const float*)d_in[7];

  const int n_nodes = in_sizes[0] / NFEAT;
  const int n_edges = in_sizes[1] / 2;
  const int total = n_nodes * NFEAT;

  // Workspace layout (floats)
  float* h0 = (float*)d_ws;                      // n_nodes*128
  float* h1 = h0 + (size_t)total;                // n_nodes*128
  float* stats = h1 + (size_t)total;             // sum[128], sumsq[128]
  float* sum = stats;
  float* sumsq = stats + 128;
  float* scale = stats + 256;
  float* shift = stats + 384;

  const int* src = edge;
  const int* dst = edge + n_edges;

  // 1. h0 = x, zero stats
  {
    int n4 = total / 4;
    int g = (n4 + 255) / 256;
    k_init<<<g, 256, 0, stream>>>((const float4*)x, (float4*)h0, stats, n4);
  }
  // 2. h0 += scatter(x[src] -> dst)
  {
    int g = (n_edges + 7) / 8;
    k_scatter<<<g, 256, 0, stream>>>(src, dst, x, h0, n_edges);
  }
  // 3. h1 = relu(h0 @ W1 + b1)
  {
    int g = (n_nodes + 15) / 16;
    k_gemm<0><<<g, 256, 0, stream>>>(h0, W1, b1, nullptr, nullptr, h1, n_nodes);
  }
  // 4. per-feature sum/sumsq
  k_stats<<<512, 256, 0, stream>>>(h1, sum, sumsq, total);
  // 5. scale/shift
  k_bnfinal<<<1, 128, 0, stream>>>(sum, sumsq, gamma, beta, scale, shift,
                                   1.0f / (float)n_nodes);
  // 6. out = (h1*scale + shift) @ W2 + b2
  {
    int g = (n_nodes + 15) / 16;
    k_gemm<1><<<g, 256, 0, stream>>>(h1, W2, b2, scale, shift, (float*)d_out,
                                     n_nodes);
  }
}